// Attention_29274497090240
// MI455X (gfx1250) — compile-verified
//
#include <hip/hip_runtime.h>
#include <hip/hip_bf16.h>
#include <math.h>

typedef __attribute__((ext_vector_type(2))) float v2f;
typedef __attribute__((ext_vector_type(4))) float v4f;
typedef __attribute__((ext_vector_type(8))) float v8f;

#define QDIM  4096
#define LROWS 8192

// ---------------------------------------------------------------------------
// Split-K GEMV via V_WMMA_F32_16X16X4_F32.
// Each wave owns 16 rows of M; the vector x sits in column 0 of B, so the
// per-row dot products accumulate in column 0 of the 16x16 f32 accumulator.
// K order is permuted so each lane's A fetch is two contiguous b128 loads:
//   WMMA0 consumes global k = {k0,k0+1} (kh=0 lanes) and {k0+4,k0+5} (kh=1)
//   WMMA1 -> {k0+2,k0+3} / {k0+6,k0+7};  WMMA2/3 same with +8.
// Four independent accumulator chains give WMMA ILP.
// y_part[seg*R + row] = sum_{k in seg} M[row,k] * x[k]
// ---------------------------------------------------------------------------
__global__ void __launch_bounds__(256) gemv16_wmma_f32(
    const float* __restrict__ M, const float* __restrict__ x,
    float* __restrict__ y_part, int C, int segC)
{
    extern __shared__ float sx[];          // staged x segment (segC floats)
    const int tid   = threadIdx.x;
    const int kbase = blockIdx.y * segC;

    {   // coalesced stage of the vector segment into LDS
        const v4f* xs = (const v4f*)(x + kbase);
        v4f*       sd = (v4f*)sx;
        for (int i = tid; i < (segC >> 2); i += 256) sd[i] = xs[i];
    }
    __syncthreads();

    const int lane    = tid & 31;
    const int m       = lane & 15;      // row within the 16-row tile / N index
    const int kh      = lane >> 4;      // which K half this lane supplies
    const int wave    = tid >> 5;       // 8 waves per block (wave32)
    const int R       = (int)gridDim.x * 128;
    const int rowbase = (int)blockIdx.x * 128 + wave * 16;
    const float* Arow = M + (size_t)(rowbase + m) * (size_t)C + kbase;
    const float  bm   = (m == 0) ? 1.0f : 0.0f;   // select column 0 of B, no branch

    v8f acc0 = {}, acc1 = {}, acc2 = {}, acc3 = {};
    for (int k0 = 0; k0 < segC; k0 += 16) {
        const int off_lo = k0 + kh * 4;
        const int off_hi = k0 + 8 + kh * 4;
        v4f a_lo = *(const v4f*)(Arow + off_lo);   // global_load_b128, contiguous
        v4f a_hi = *(const v4f*)(Arow + off_hi);
        v4f x_lo = *(const v4f*)(sx + off_lo);     // LDS broadcast reads
        v4f x_hi = *(const v4f*)(sx + off_hi);

        v2f a0 = {a_lo.x, a_lo.y}, a1 = {a_lo.z, a_lo.w};
        v2f a2 = {a_hi.x, a_hi.y}, a3 = {a_hi.z, a_hi.w};
        v2f b0 = {x_lo.x * bm, x_lo.y * bm}, b1 = {x_lo.z * bm, x_lo.w * bm};
        v2f b2 = {x_hi.x * bm, x_hi.y * bm}, b3 = {x_hi.z * bm, x_hi.w * bm};

        acc0 = __builtin_amdgcn_wmma_f32_16x16x4_f32(false, a0, false, b0, (short)0, acc0, false, false);
        acc1 = __builtin_amdgcn_wmma_f32_16x16x4_f32(false, a1, false, b1, (short)0, acc1, false, false);
        acc2 = __builtin_amdgcn_wmma_f32_16x16x4_f32(false, a2, false, b2, (short)0, acc2, false, false);
        acc3 = __builtin_amdgcn_wmma_f32_16x16x4_f32(false, a3, false, b3, (short)0, acc3, false, false);
    }
    v8f t = acc0 + acc1 + acc2 + acc3;

    // Column 0 of C: lane 0 holds rows M=0..7 in v0..7, lane 16 holds M=8..15.
    if (m == 0) {
        float* dst = y_part + (size_t)blockIdx.y * R + rowbase + kh * 8;
        #pragma unroll
        for (int r = 0; r < 8; ++r) dst[r] = t[r];
    }
}

// y[i] = sum_s p[s*R + i]  — deterministic split-K reduction
__global__ void __launch_bounds__(256) reduce_part(
    const float* __restrict__ p, float* __restrict__ y, int R, int nseg)
{
    int i = blockIdx.x * 256 + threadIdx.x;
    if (i < R) {
        float s = 0.0f;
        for (int j = 0; j < nseg; ++j) s += p[(size_t)j * R + i];
        y[i] = s;
    }
}

// Softmax over 8192 scores (in place: scores -> attention) and copy query
// into gamma[0..Q). Single 1024-thread block.
__global__ void __launch_bounds__(1024) softmax_fuse(
    float* __restrict__ score, const float* __restrict__ query,
    float* __restrict__ gamma)
{
    __shared__ float red[1024];
    const int tid = threadIdx.x;

    float m = -INFINITY;
    #pragma unroll
    for (int j = 0; j < LROWS / 1024; ++j) m = fmaxf(m, score[tid + j * 1024]);
    red[tid] = m;
    __syncthreads();
    for (int s = 512; s > 0; s >>= 1) {
        if (tid < s) red[tid] = fmaxf(red[tid], red[tid + s]);
        __syncthreads();
    }
    const float gmax = red[0];
    __syncthreads();

    float e[LROWS / 1024];
    float sum = 0.0f;
    #pragma unroll
    for (int j = 0; j < LROWS / 1024; ++j) {
        e[j] = expf(score[tid + j * 1024] - gmax);
        sum += e[j];
    }
    red[tid] = sum;
    __syncthreads();
    for (int s = 512; s > 0; s >>= 1) {
        if (tid < s) red[tid] += red[tid + s];
        __syncthreads();
    }
    const float inv = 1.0f / red[0];
    #pragma unroll
    for (int j = 0; j < LROWS / 1024; ++j) score[tid + j * 1024] = e[j] * inv;

    for (int i = tid; i < QDIM; i += 1024) gamma[i] = query[i];
}

// Split-L weighted column sum: p[seg*Q + q] = sum_{l in seg} att[l]*ctx[l][q].
// Threads map to adjacent columns -> fully coalesced b32 loads; context is
// L2-resident after the score pass (128 MB < 192 MB L2).
__global__ void __launch_bounds__(256) wsum_part(
    const float* __restrict__ ctx, const float* __restrict__ att,
    float* __restrict__ p)
{
    __shared__ float sa[LROWS / 16];   // 512 attention weights, 2 KB
    const int tid   = threadIdx.x;
    const int lbase = blockIdx.y * (LROWS / 16);
    for (int i = tid; i < LROWS / 16; i += 256) sa[i] = att[lbase + i];
    __syncthreads();

    const int q = blockIdx.x * 256 + tid;
    const float* c = ctx + (size_t)lbase * QDIM + q;
    float acc = 0.0f;
    #pragma unroll 8
    for (int l = 0; l < LROWS / 16; ++l)
        acc = fmaf(sa[l], c[(size_t)l * QDIM], acc);
    p[(size_t)blockIdx.y * QDIM + q] = acc;
}

// ---------------------------------------------------------------------------
extern "C" void kernel_launch(void* const* d_in, const int* in_sizes, int n_in,
                              void* d_out, int out_size, void* d_ws, size_t ws_size,
                              hipStream_t stream)
{
    (void)in_sizes; (void)n_in; (void)out_size; (void)ws_size;
    const float* query = (const float*)d_in[0];          // [Q]
    const float* ctx   = (const float*)d_in[1];          // [L, Q]
    const float* Kw    = (const float*)d_in[2];          // [Q, 2Q]
    float* out = (float*)d_out;                          // [Q]

    float* ws    = (float*)d_ws;
    float* score = ws;                     // [L]        scores -> attention
    float* gamma = ws + LROWS;             // [2Q]       [query ; s_t]
    float* part  = ws + LROWS + 2 * QDIM;  // [<=64K]    split partials (reused)

    // 1) score = ctx @ query   (R=8192, C=4096, 8-way split-K)
    gemv16_wmma_f32<<<dim3(LROWS / 128, 8), 256, (QDIM / 8) * sizeof(float), stream>>>(
        ctx, query, part, QDIM, QDIM / 8);
    reduce_part<<<LROWS / 256, 256, 0, stream>>>(part, score, LROWS, 8);

    // 2) attention = softmax(score); gamma[0..Q) = query
    softmax_fuse<<<1, 1024, 0, stream>>>(score, query, gamma);

    // 3) s_t = attention @ ctx  (16-way split-L), gamma[Q..2Q) = s_t
    wsum_part<<<dim3(QDIM / 256, 16), 256, 0, stream>>>(ctx, score, part);
    reduce_part<<<QDIM / 256, 256, 0, stream>>>(part, gamma + QDIM, QDIM, 16);

    // 4) out = K_w @ gamma     (R=4096, C=8192, 8-way split-K)
    gemv16_wmma_f32<<<dim3(QDIM / 128, 8), 256, (2 * QDIM / 8) * sizeof(float), stream>>>(
        Kw, gamma, part, 2 * QDIM, 2 * QDIM / 8);
    reduce_part<<<QDIM / 256, 256, 0, stream>>>(part, out, QDIM, 8);
}